// FilterDetections_21878563406407
// MI455X (gfx1250) — compile-verified
//
#include <hip/hip_runtime.h>
#include <hip/hip_bf16.h>

// ---------------- problem constants (match reference) ----------------
#define BATCH     8
#define NANCH     49104
#define NCLS      90
#define TOPK      1000     // PRE_NMS_TOPK
#define CAP       1024     // padded candidate capacity
#define SEL       300      // MAX_DETECTIONS
#define CHUNK     2048     // anchors per TDM chunk (8KB fp32)
#define NBINS     4096     // linear score histogram bins
#define EQCAP     1024     // boundary-bin overflow list capacity
#define THRVAL    0.01f
#define IOUTHR    0.5f
#define NEGV      (-1.0e9f)
#define NEGHALF   (-5.0e8f)

// ---------------- CDNA5 TDM plumbing (device pass only) ----------------
// The TDM builtins exist only in the device compilation; the host pass
// merely needs the code to parse. The #error probe is kept for the DEVICE
// pass: a successful compile proves tensor_load_to_lds was emitted.
#if defined(__HIP_DEVICE_COMPILE__)
#  if !(__has_builtin(__builtin_amdgcn_tensor_load_to_lds) && __has_builtin(__builtin_amdgcn_s_wait_tensorcnt))
#    error "CDNA5 TDM builtins unavailable in DEVICE compilation (probe)"
#  endif
#  define WAIT_TENSORCNT(n) __builtin_amdgcn_s_wait_tensorcnt((short)(n))
#else
#  define WAIT_TENSORCNT(n) ((void)0)
#endif

typedef unsigned int v4u __attribute__((ext_vector_type(4)));
typedef int          v4i __attribute__((ext_vector_type(4)));
typedef int          v8i __attribute__((ext_vector_type(8)));

// Gather `rows` fp32 elements starting at gbase with an element stride of
// `stride_elems`, compacted into LDS at lds_dst.  2-D descriptor:
//   tile = 1 x rows, tensor_dim0_stride = stride_elems, data_size = 4B.
__device__ __forceinline__ void tdm_gather_column(void* lds_dst, const float* gbase,
                                                  int rows, int stride_elems) {
#if defined(__HIP_DEVICE_COMPILE__)
  unsigned lds_addr = (unsigned)(unsigned long long)(size_t)lds_dst; // low 32 bits of flat = LDS byte addr
  unsigned long long ga = (unsigned long long)(size_t)gbase;

  v4u g0;
  g0[0] = 1u;                                                  // count=1 valid descriptor
  g0[1] = lds_addr;                                            // lds_addr[31:0]
  g0[2] = (unsigned)(ga & 0xFFFFFFFFull);                      // global_addr[31:0]
  g0[3] = (unsigned)((ga >> 32) & 0x01FFFFFFull) | (2u << 30); // global_addr[56:32] | type=2

  v8i g1;
  g1[0] = 0x00020000;                                          // data_size=2 (4B), no multicast
  g1[1] = 0x00010000;                                          // tensor_dim0=1 (bits 79:48 low half)
  g1[2] = (int)(((unsigned)rows & 0xFFFFu) << 16);             // tensor_dim1[15:0] = rows
  g1[3] = 0x00010000;                                          // tile_dim0=1
  g1[4] = rows;                                                // tile_dim1=rows, tile_dim2=0
  g1[5] = stride_elems;                                        // tensor_dim0_stride[31:0]
  g1[6] = 0;                                                   // stride0 hi, stride1 lo
  g1[7] = 0;                                                   // stride1 hi

  v4i gz = {0, 0, 0, 0};
#if defined(__clang_major__) && __clang_major__ >= 23
  v8i gz8 = {0, 0, 0, 0, 0, 0, 0, 0};
  __builtin_amdgcn_tensor_load_to_lds(g0, g1, gz, gz, gz8, 0);
#else
  __builtin_amdgcn_tensor_load_to_lds(g0, g1, gz, gz, 0);
#endif
#else
  (void)lds_dst; (void)gbase; (void)rows; (void)stride_elems;
#endif
}

// ---------------- helpers ----------------
__device__ __forceinline__ unsigned order_f(float f) {
  unsigned u = __float_as_uint(f);
  return (u & 0x80000000u) ? ~u : (u | 0x80000000u);  // monotone float->uint
}
__device__ __forceinline__ unsigned long long pack_key(float s, unsigned idx) {
  // max(key) == max score, ties -> lowest idx (matches jnp.argmax / top_k)
  return ((unsigned long long)order_f(s) << 32) | (unsigned long long)(0xFFFFFFFFu - idx);
}
__device__ __forceinline__ unsigned long long umax64(unsigned long long a, unsigned long long b) {
  return a > b ? a : b;
}

// =====================================================================
// Kernel 1: one block per (batch, class) pair.
//   Phase A: TDM-stream the class score column through LDS chunks,
//            histogram-select the exact top-1000 candidates.
//   Phase B: gather candidate boxes, greedy NMS for 300 selections.
// =====================================================================
__global__ __launch_bounds__(256) void fd_nms_kernel(const float* __restrict__ boxes,
                                                     const float* __restrict__ cls,
                                                     float* __restrict__ selS,
                                                     float* __restrict__ selB) {
  const int pair = blockIdx.x;
  const int b = pair / NCLS;
  const int c = pair % NCLS;
  const int tid = threadIdx.x;

  __shared__ float s_cand_score[CAP];
  __shared__ int   s_cand_idx[CAP];
  __shared__ unsigned long long s_red[256];
  __shared__ int   s_ctl[8]; // 0:cntHigh 1:cntEq 2:T 3:need0 4:nh 5:need 6:total 7:ne
  __shared__ __align__(16) unsigned char s_blob[40960];

  // Phase-A carving of blob
  unsigned* hist  = (unsigned*)s_blob;                  // 16KB
  float* chunk0   = (float*)(s_blob + 16384);           // 8KB
  float* chunk1   = (float*)(s_blob + 24576);           // 8KB
  float* eqv      = (float*)(s_blob + 32768);           // 4KB
  int*   eqi      = (int*)(s_blob + 36864);             // 4KB
  // Phase-B carving (reuses blob after selection is done)
  float* bx1   = (float*)(s_blob);
  float* by1   = (float*)(s_blob + 4096);
  float* bx2   = (float*)(s_blob + 8192);
  float* by2   = (float*)(s_blob + 12288);
  float* barea = (float*)(s_blob + 16384);

  for (int i = tid; i < NBINS; i += 256) hist[i] = 0u;
  if (tid < 8) s_ctl[tid] = 0;
  __syncthreads();

  const int nch = (NANCH + CHUNK - 1) / CHUNK;
  const float* colbase = cls + ((size_t)b * NANCH) * NCLS + c;

  // -------- Pass 1: histogram (TDM double-buffered stream) --------
  if (tid == 0) {
    int r0 = NANCH < CHUNK ? NANCH : CHUNK;
    tdm_gather_column(chunk0, colbase, r0, NCLS);
  }
  for (int k = 0; k < nch; ++k) {
    const int i0 = k * CHUNK;
    const int rows = (NANCH - i0) < CHUNK ? (NANCH - i0) : CHUNK;
    float* cur = (k & 1) ? chunk1 : chunk0;
    if (tid == 0) {
      if (k + 1 < nch) {
        int i1 = (k + 1) * CHUNK;
        int r1 = (NANCH - i1) < CHUNK ? (NANCH - i1) : CHUNK;
        tdm_gather_column(((k + 1) & 1) ? chunk1 : chunk0, colbase + (size_t)i1 * NCLS, r1, NCLS);
        WAIT_TENSORCNT(1); // chunk k complete (in-order per wave)
      } else {
        WAIT_TENSORCNT(0);
      }
    }
    __syncthreads();
    for (int j = tid; j < rows; j += 256) {
      float s = cur[j];
      if (s > THRVAL) {
        int bin = (int)(s * (float)NBINS);
        bin = bin < 0 ? 0 : (bin > NBINS - 1 ? NBINS - 1 : bin);
        atomicAdd(&hist[bin], 1u);
      }
    }
    __syncthreads(); // chunk consumed before its buffer is refilled
  }

  // -------- cutoff bin --------
  if (tid == 0) {
    int acc = 0, t;
    for (t = NBINS - 1; t >= 0; --t) {
      int h = (int)hist[t];
      if (acc + h >= TOPK) break;
      acc += h;
    }
    s_ctl[2] = t;                       // boundary bin (-1 => take everything)
    s_ctl[3] = (t >= 0) ? (TOPK - acc) : 0;
    s_ctl[0] = 0; s_ctl[1] = 0;
  }
  __syncthreads();
  const int Tbin = s_ctl[2];

  // -------- Pass 2: collect candidates --------
  if (tid == 0) {
    int r0 = NANCH < CHUNK ? NANCH : CHUNK;
    tdm_gather_column(chunk0, colbase, r0, NCLS);
  }
  for (int k = 0; k < nch; ++k) {
    const int i0 = k * CHUNK;
    const int rows = (NANCH - i0) < CHUNK ? (NANCH - i0) : CHUNK;
    float* cur = (k & 1) ? chunk1 : chunk0;
    if (tid == 0) {
      if (k + 1 < nch) {
        int i1 = (k + 1) * CHUNK;
        int r1 = (NANCH - i1) < CHUNK ? (NANCH - i1) : CHUNK;
        tdm_gather_column(((k + 1) & 1) ? chunk1 : chunk0, colbase + (size_t)i1 * NCLS, r1, NCLS);
        WAIT_TENSORCNT(1);
      } else {
        WAIT_TENSORCNT(0);
      }
    }
    __syncthreads();
    for (int j = tid; j < rows; j += 256) {
      float s = cur[j];
      if (s > THRVAL) {
        int bin = (int)(s * (float)NBINS);
        bin = bin < 0 ? 0 : (bin > NBINS - 1 ? NBINS - 1 : bin);
        if (Tbin < 0 || bin > Tbin) {
          int p = atomicAdd(&s_ctl[0], 1);
          if (p < CAP) { s_cand_score[p] = s; s_cand_idx[p] = i0 + j; }
        } else if (bin == Tbin) {
          int p = atomicAdd(&s_ctl[1], 1);
          if (p < EQCAP) { eqv[p] = s; eqi[p] = i0 + j; }
        }
      }
    }
    __syncthreads();
  }

  if (tid == 0) {
    int nh = s_ctl[0]; nh = nh > CAP ? CAP : nh;
    int ne = s_ctl[1]; ne = ne > EQCAP ? EQCAP : ne;
    int need = s_ctl[3]; need = need > ne ? ne : need;
    if (Tbin < 0) need = 0;
    int total = nh + need; total = total > TOPK ? TOPK : total;
    s_ctl[4] = nh; s_ctl[5] = need; s_ctl[6] = total; s_ctl[7] = ne;
  }
  __syncthreads();
  const int nh = s_ctl[4], need = s_ctl[5], total = s_ctl[6], ne = s_ctl[7];

  // Exact boundary-bin fill: rank-select top `need` of the eq list by
  // (score desc, anchor-index asc) -- reproduces top_k tie-breaks.
  for (int t2 = tid; t2 < ne; t2 += 256) {
    const unsigned long long kt = pack_key(eqv[t2], (unsigned)eqi[t2]);
    int rank = 0;
    for (int u = 0; u < ne; ++u)
      rank += (pack_key(eqv[u], (unsigned)eqi[u]) > kt) ? 1 : 0;
    if (rank < need) { s_cand_score[nh + rank] = eqv[t2]; s_cand_idx[nh + rank] = eqi[t2]; }
  }
  __syncthreads();
  for (int i = tid; i < CAP; i += 256) {
    if (i >= total) { s_cand_score[i] = NEGV; s_cand_idx[i] = 0; }
  }
  __syncthreads();

  // -------- Phase B: candidate boxes (blob reused) --------
  for (int i = tid; i < CAP; i += 256) {
    if (i < total) {
      const float4 bb = ((const float4*)boxes)[(size_t)b * NANCH + s_cand_idx[i]];
      bx1[i] = bb.x; by1[i] = bb.y; bx2[i] = bb.z; by2[i] = bb.w;
      barea[i] = (bb.z - bb.x) * (bb.w - bb.y);
    } else {
      bx1[i] = 0.f; by1[i] = 0.f; bx2[i] = 0.f; by2[i] = 0.f; barea[i] = 0.f;
    }
  }
  __syncthreads();

  // -------- greedy NMS: 300 selections --------
  float* selS_p = selS + (size_t)pair * SEL;
  float* selB_p = selB + (size_t)pair * SEL * 4;
  for (int it = 0; it < SEL; ++it) {
    unsigned long long best = 0ull;
    for (int i = tid; i < CAP; i += 256)
      best = umax64(best, pack_key(s_cand_score[i], (unsigned)i));
    s_red[tid] = best;
    __syncthreads();
    for (int off = 128; off; off >>= 1) {
      if (tid < off) s_red[tid] = umax64(s_red[tid], s_red[tid + off]);
      __syncthreads();
    }
    const unsigned long long top = s_red[0];
    const int j = (int)(0xFFFFFFFFu - (unsigned)top);
    const float bs = s_cand_score[j];
    const bool valid = bs > NEGHALF;  // uniform

    if (!valid) {
      for (int m = it + tid; m < SEL; m += 256) {
        selS_p[m] = NEGV;
        selB_p[(size_t)m * 4 + 0] = -1.f; selB_p[(size_t)m * 4 + 1] = -1.f;
        selB_p[(size_t)m * 4 + 2] = -1.f; selB_p[(size_t)m * 4 + 3] = -1.f;
      }
      break;
    }

    const float ax1 = bx1[j], ay1 = by1[j], ax2 = bx2[j], ay2 = by2[j], aA = barea[j];
    for (int i = tid; i < CAP; i += 256) {
      float ix1 = fmaxf(ax1, bx1[i]);
      float iy1 = fmaxf(ay1, by1[i]);
      float ix2 = fminf(ax2, bx2[i]);
      float iy2 = fminf(ay2, by2[i]);
      float inter = fmaxf(ix2 - ix1, 0.f) * fmaxf(iy2 - iy1, 0.f);
      float iou = inter / fmaxf(aA + barea[i] - inter, 1e-8f);
      if (iou > IOUTHR) s_cand_score[i] = NEGV;  // includes the selected box
    }
    if (tid == 0) {
      selS_p[it] = bs;
      selB_p[(size_t)it * 4 + 0] = ax1; selB_p[(size_t)it * 4 + 1] = ay1;
      selB_p[(size_t)it * 4 + 2] = ax2; selB_p[(size_t)it * 4 + 3] = ay2;
    }
    __syncthreads();
  }
}

// =====================================================================
// Kernel 2: one block per batch. Per-class selection lists are sorted
// descending by construction -> 90-way sorted merge, top 300,
// tie-break lowest class (== lowest flat index, matching top_k).
// =====================================================================
__global__ __launch_bounds__(128) void fd_merge_kernel(const float* __restrict__ selS,
                                                       const float* __restrict__ selB,
                                                       float* __restrict__ out_boxes,
                                                       float* __restrict__ out_scores,
                                                       int* __restrict__ out_labels) {
  const int b = blockIdx.x;
  const int tid = threadIdx.x;
  __shared__ int head[NCLS];
  __shared__ unsigned long long red[128];

  if (tid < NCLS) head[tid] = 0;
  __syncthreads();

  for (int m = 0; m < SEL; ++m) {
    unsigned long long key = 0ull;
    if (tid < NCLS) {
      int p = head[tid];
      float s = (p < SEL) ? selS[((size_t)b * NCLS + tid) * SEL + p] : NEGV;
      key = pack_key(s, (unsigned)tid);
    }
    red[tid] = key;
    __syncthreads();
    for (int off = 64; off; off >>= 1) {
      if (tid < off) red[tid] = umax64(red[tid], red[tid + off]);
      __syncthreads();
    }
    if (tid == 0) {
      const unsigned long long top = red[0];
      const int c = (int)(0xFFFFFFFFu - (unsigned)top);
      const int p = head[c];
      const float s = (p < SEL) ? selS[((size_t)b * NCLS + c) * SEL + p] : NEGV;
      const size_t o = (size_t)b * SEL + m;
      if (s > NEGHALF) {
        const size_t src = (((size_t)b * NCLS + c) * SEL + p) * 4;
        out_boxes[o * 4 + 0] = selB[src + 0];
        out_boxes[o * 4 + 1] = selB[src + 1];
        out_boxes[o * 4 + 2] = selB[src + 2];
        out_boxes[o * 4 + 3] = selB[src + 3];
        out_scores[o] = s;
        out_labels[o] = c;
        head[c] = p + 1;
      } else {
        out_boxes[o * 4 + 0] = -1.f; out_boxes[o * 4 + 1] = -1.f;
        out_boxes[o * 4 + 2] = -1.f; out_boxes[o * 4 + 3] = -1.f;
        out_scores[o] = -1.f;
        out_labels[o] = -1;
      }
    }
    __syncthreads();
  }
}

// =====================================================================
extern "C" void kernel_launch(void* const* d_in, const int* in_sizes, int n_in,
                              void* d_out, int out_size, void* d_ws, size_t ws_size,
                              hipStream_t stream) {
  const float* boxes = (const float*)d_in[0];          // [B,N,4]
  const float* cls   = (const float*)d_in[1];          // [B,N,C]

  float* selS = (float*)d_ws;                          // [B*C, 300]
  float* selB = selS + (size_t)BATCH * NCLS * SEL;     // [B*C, 300, 4]

  float* out_boxes  = (float*)d_out;                   // [B,300,4]
  float* out_scores = out_boxes + (size_t)BATCH * SEL * 4;
  int*   out_labels = (int*)(out_scores + (size_t)BATCH * SEL);

  fd_nms_kernel<<<BATCH * NCLS, 256, 0, stream>>>(boxes, cls, selS, selB);
  fd_merge_kernel<<<BATCH, 128, 0, stream>>>(selS, selB, out_boxes, out_scores, out_labels);
}